// UpdateNode_15496242004735
// MI455X (gfx1250) — compile-verified
//
#include <hip/hip_runtime.h>
#include <hip/hip_bf16.h>
#include <math.h>

#define D 128
#define H 512
#define EPSV 1e-5f

typedef float v2f __attribute__((ext_vector_type(2)));
typedef float v8f __attribute__((ext_vector_type(8)));

#if defined(__has_builtin)
#if __has_builtin(__builtin_amdgcn_global_load_async_to_lds_b32) && \
    __has_builtin(__builtin_amdgcn_s_wait_asynccnt)
#define HAVE_ASYNC_LDS 1
#endif
#endif
#ifndef HAVE_ASYNC_LDS
#define HAVE_ASYNC_LDS 0
#endif

#define AS1 __attribute__((address_space(1)))
#define AS3 __attribute__((address_space(3)))

#if HAVE_ASYNC_LDS
// DMA one dword global -> LDS (no VGPR destination, tracked by ASYNCcnt).
// LDS aperture maps addr[31:0] to the LDS byte offset (ISA 10.2), so casting
// the flat shared-pointer through uintptr_t yields a valid as(3) address.
__device__ __forceinline__ void async_cp_f32(const float* g, float* l) {
  __builtin_amdgcn_global_load_async_to_lds_b32(
      (AS1 int*)(uintptr_t)g, (AS3 int*)(uintptr_t)l, 0, 0);
}
#endif

__device__ __forceinline__ v8f wmma_f32(v2f a, v2f b, v8f c) {
  // D = A(16x4, f32) * B(4x16, f32) + C(16x16, f32)
  return __builtin_amdgcn_wmma_f32_16x16x4_f32(false, a, false, b, (short)0, c,
                                               false, false);
}

// ---------------- k0: zero reduction scratch ----------------
__global__ void k0_zero(float* ws, int n) {
  int i = blockIdx.x * blockDim.x + threadIdx.x;
  if (i < n) ws[i] = 0.0f;
}

// ---------------- k1: BN1 column stats ----------------
__global__ __launch_bounds__(256) void k1_bn_stats(
    const float* __restrict__ x, float* __restrict__ sum,
    float* __restrict__ sq, int rowsPerBlock) {
  __shared__ float sS[D], sQ[D];
  int t = threadIdx.x;
  if (t < D) { sS[t] = 0.f; sQ[t] = 0.f; }
  __syncthreads();
  int base = blockIdx.x * rowsPerBlock;
  int subrow = t >> 2;
  int colb = (t & 3) * 32;
  float ls[32], lq[32];
#pragma unroll
  for (int c = 0; c < 32; ++c) { ls[c] = 0.f; lq[c] = 0.f; }
  int iters = rowsPerBlock >> 6;  // 64 rows per sweep
  for (int i = 0; i < iters; ++i) {
    const float* row = x + (size_t)(base + i * 64 + subrow) * D + colb;
#pragma unroll
    for (int c = 0; c < 32; c += 4) {
      float4 v = *(const float4*)(row + c);
      ls[c + 0] += v.x; lq[c + 0] += v.x * v.x;
      ls[c + 1] += v.y; lq[c + 1] += v.y * v.y;
      ls[c + 2] += v.z; lq[c + 2] += v.z * v.z;
      ls[c + 3] += v.w; lq[c + 3] += v.w * v.w;
    }
  }
#pragma unroll
  for (int c = 0; c < 32; ++c) {
    atomicAdd(&sS[colb + c], ls[c]);
    atomicAdd(&sQ[colb + c], lq[c]);
  }
  __syncthreads();
  if (t < D) {
    atomicAdd(&sum[t], sS[t]);
    atomicAdd(&sq[t], sQ[t]);
  }
}

// ---------------- k1b: fold BN1 stats into affine constants ----------------
__global__ void k1b_fold(const float* __restrict__ sum, const float* __restrict__ sq,
                         const float* __restrict__ g, const float* __restrict__ b,
                         float* __restrict__ bnA, float* __restrict__ bnB, float invN) {
  int c = threadIdx.x;
  float m = sum[c] * invN;
  float v = sq[c] * invN - m * m;
  float A = rsqrtf(v + EPSV) * g[c];
  bnA[c] = A;
  bnB[c] = b[c] - m * A;
}

// ---------------- k2: fused BN1 -> GEMM1 -> ReLU -> GEMM2, BN2 stats ----------------
// Double-buffered weight tiles staged by the async global->LDS DMA (ASYNCcnt),
// drained with s_wait_asynccnt before the single per-iteration barrier.
__global__ __launch_bounds__(256) void k2_fused_mlp(
    const float* __restrict__ hV, const float* __restrict__ w1,
    const float* __restrict__ b1, const float* __restrict__ w2,
    const float* __restrict__ b2, const float* __restrict__ bnA,
    const float* __restrict__ bnB, float* __restrict__ dh_raw,
    float* __restrict__ bn2_sum, float* __restrict__ bn2_sq) {
  __shared__ float sBnA[D], sBnB[D];
  __shared__ float sW1T[2][16][130];   // [buf][local col][k]   (W1 block, transposed)
  __shared__ float sW2T[2][D][18];     // [buf][col][local k]   (W2 block, transposed)
  __shared__ float sY[8][16][17];      // per-wave Y tile staging (C->A relayout)
  __shared__ float sS2[D], sQ2[D];

  const int t = threadIdx.x;
  const int lane = t & 31;
  const int wv = t >> 5;
  const int l16 = lane & 15;
  const int lhi = lane >> 4;  // 0 or 1
  const int rowbase = blockIdx.x * 128 + wv * 16;
  const int rowg = rowbase + l16;

#if !HAVE_ASYNC_LDS
  float rw1[8], rw2[8];
#endif

  if (t < D) { sBnA[t] = bnA[t]; sBnB[t] = bnB[t]; sS2[t] = 0.f; sQ2[t] = 0.f; }

  // prologue: stage weight tile nt=0 into LDS buffer 0
#if HAVE_ASYNC_LDS
#pragma unroll
  for (int j = 0; j < 8; ++j) {
    int i1 = t + 256 * j;
    async_cp_f32(w1 + (size_t)(i1 >> 4) * H + (i1 & 15),
                 &sW1T[0][i1 & 15][i1 >> 4]);
    async_cp_f32(w2 + (size_t)(i1 >> 7) * D + (i1 & 127),
                 &sW2T[0][i1 & 127][i1 >> 7]);
  }
#else
#pragma unroll
  for (int j = 0; j < 8; ++j) {
    int i1 = t + 256 * j;
    rw1[j] = w1[(size_t)(i1 >> 4) * H + (i1 & 15)];
    rw2[j] = w2[(size_t)(i1 >> 7) * D + (i1 & 127)];
  }
#endif
  __syncthreads();  // sBnA visible

  // X as A-fragments (BN1 applied). lanes 0-15: K = 4kk+{0,1}; lanes 16-31: 4kk+{2,3}
  v2f ax[32];
  const float* hrow = hV + (size_t)rowg * D;
#pragma unroll
  for (int kk = 0; kk < 32; ++kk) {
    int c0 = kk * 4 + lhi * 2;
    ax[kk].x = hrow[c0] * sBnA[c0] + sBnB[c0];
    ax[kk].y = hrow[c0 + 1] * sBnA[c0 + 1] + sBnB[c0 + 1];
  }

#if !HAVE_ASYNC_LDS
#pragma unroll
  for (int j = 0; j < 8; ++j) {
    int i1 = t + 256 * j;
    sW1T[0][i1 & 15][i1 >> 4] = rw1[j];
    sW2T[0][i1 & 127][i1 >> 7] = rw2[j];
  }
#endif

  v8f zacc[8];
#pragma unroll
  for (int n2 = 0; n2 < 8; ++n2)
    zacc[n2] = (v8f){0.f, 0.f, 0.f, 0.f, 0.f, 0.f, 0.f, 0.f};

#if HAVE_ASYNC_LDS
  __builtin_amdgcn_s_wait_asynccnt(0);
#endif
  __syncthreads();  // buffer 0 visible

  for (int nt = 0; nt < 32; ++nt) {
    const int buf = nt & 1;
    // stage the NEXT tile into the other buffer (overlaps with WMMA below)
    if (nt + 1 < 32) {
#if HAVE_ASYNC_LDS
#pragma unroll
      for (int j = 0; j < 8; ++j) {
        int i1 = t + 256 * j;
        async_cp_f32(w1 + (size_t)(i1 >> 4) * H + (nt + 1) * 16 + (i1 & 15),
                     &sW1T[buf ^ 1][i1 & 15][i1 >> 4]);
        async_cp_f32(w2 + (size_t)((nt + 1) * 16 + (i1 >> 7)) * D + (i1 & 127),
                     &sW2T[buf ^ 1][i1 & 127][i1 >> 7]);
      }
#else
#pragma unroll
      for (int j = 0; j < 8; ++j) {
        int i1 = t + 256 * j;
        rw1[j] = w1[(size_t)(i1 >> 4) * H + (nt + 1) * 16 + (i1 & 15)];
        rw2[j] = w2[(size_t)((nt + 1) * 16 + (i1 >> 7)) * D + (i1 & 127)];
      }
#endif
    }

    // GEMM1 tile: Y[16 x 16] for cols nt*16..+16, K=128
    // (accumulator chaining D->C is hazard-free per ISA 7.12.1)
    v8f y = (v8f){0.f, 0.f, 0.f, 0.f, 0.f, 0.f, 0.f, 0.f};
#pragma unroll
    for (int kk = 0; kk < 32; ++kk) {
      int k = kk * 4 + lhi * 2;
      v2f bf;
      bf.x = sW1T[buf][l16][k];
      bf.y = sW1T[buf][l16][k + 1];
      y = wmma_f32(ax[kk], bf, y);
    }
    // bias + ReLU in accumulator layout (col = nt*16 + l16 for all 8 comps)
    float bb = b1[nt * 16 + l16];
#pragma unroll
    for (int r = 0; r < 8; ++r) {
      float v = y[r] + bb;
      y[r] = v > 0.f ? v : 0.f;
    }
    // relayout C -> A via per-wave LDS staging
#pragma unroll
    for (int r = 0; r < 8; ++r) sY[wv][r + 8 * lhi][l16] = y[r];
    asm volatile("s_wait_dscnt 0" ::: "memory");  // same-wave cross-lane reload
    v2f a2[4];
#pragma unroll
    for (int kk = 0; kk < 4; ++kk) {
      int k = kk * 4 + lhi * 2;
      a2[kk].x = sY[wv][l16][k];
      a2[kk].y = sY[wv][l16][k + 1];
    }
    // GEMM2 partial: Z[16 x 128] += Ytile @ W2[nt*16..+16, :]
#pragma unroll
    for (int n2 = 0; n2 < 8; ++n2) {
#pragma unroll
      for (int kk = 0; kk < 4; ++kk) {
        int k = kk * 4 + lhi * 2;
        v2f bf;
        bf.x = sW2T[buf][n2 * 16 + l16][k];
        bf.y = sW2T[buf][n2 * 16 + l16][k + 1];
        zacc[n2] = wmma_f32(a2[kk], bf, zacc[n2]);
      }
    }

#if HAVE_ASYNC_LDS
    __builtin_amdgcn_s_wait_asynccnt(0);  // our DMA into buf^1 has landed
#else
    if (nt + 1 < 32) {
#pragma unroll
      for (int j = 0; j < 8; ++j) {
        int i1 = t + 256 * j;
        sW1T[buf ^ 1][i1 & 15][i1 >> 4] = rw1[j];
        sW2T[buf ^ 1][i1 & 127][i1 >> 7] = rw2[j];
      }
    }
#endif
    __syncthreads();  // next-tile buffer visible; also fences reads of `buf`
  }

  // epilogue: +b2, store dh_raw, BN2 partial stats
#pragma unroll
  for (int n2 = 0; n2 < 8; ++n2) {
    int col = n2 * 16 + l16;
    float bb = b2[col];
    float s = 0.f, q = 0.f;
#pragma unroll
    for (int r = 0; r < 8; ++r) {
      float v = zacc[n2][r] + bb;
      int row = rowbase + r + 8 * lhi;
      dh_raw[(size_t)row * D + col] = v;
      s += v;
      q += v * v;
    }
    atomicAdd(&sS2[col], s);
    atomicAdd(&sQ2[col], q);
  }
  __syncthreads();
  if (t < D) {
    atomicAdd(&bn2_sum[t], sS2[t]);
    atomicAdd(&bn2_sq[t], sQ2[t]);
  }
}

// ---------------- k4: h = hV + BN2(dh); per-graph segment sums ----------------
__global__ __launch_bounds__(256) void k4_seg(
    const float* __restrict__ hV, const float* __restrict__ dh,
    const int* __restrict__ bid, const float* __restrict__ bn2_sum,
    const float* __restrict__ bn2_sq, const float* __restrict__ g2,
    const float* __restrict__ beta2, float* __restrict__ seg_sum,
    float* __restrict__ seg_cnt, float invN) {
  __shared__ float cA[D], cB[D], colsum[D];
  __shared__ float scnt;
  __shared__ int sB0;
  int t = threadIdx.x;
  if (t < D) {
    float m = bn2_sum[t] * invN;
    float v = bn2_sq[t] * invN - m * m;
    float A = rsqrtf(v + EPSV) * g2[t];
    cA[t] = A;
    cB[t] = beta2[t] - m * A;
    colsum[t] = 0.f;
  }
  int base = blockIdx.x * 64;
  if (t == 0) { sB0 = bid[base]; scnt = 0.f; }
  __syncthreads();
  int row = base + (t >> 2);
  int colb = (t & 3) * 32;
  int b0 = sB0;
  int bb = bid[row];
  const float* hv = hV + (size_t)row * D + colb;
  const float* dr = dh + (size_t)row * D + colb;
  float local[32];
#pragma unroll
  for (int c = 0; c < 32; c += 4) {
    float4 a = *(const float4*)(hv + c);
    float4 d = *(const float4*)(dr + c);
    local[c + 0] = a.x + d.x * cA[colb + c + 0] + cB[colb + c + 0];
    local[c + 1] = a.y + d.y * cA[colb + c + 1] + cB[colb + c + 1];
    local[c + 2] = a.z + d.z * cA[colb + c + 2] + cB[colb + c + 2];
    local[c + 3] = a.w + d.w * cA[colb + c + 3] + cB[colb + c + 3];
  }
  if (bb == b0) {
#pragma unroll
    for (int c = 0; c < 32; ++c) atomicAdd(&colsum[colb + c], local[c]);
    if ((t & 3) == 0) atomicAdd(&scnt, 1.f);
  } else {  // boundary fallback (unused for aligned segments)
#pragma unroll
    for (int c = 0; c < 32; ++c)
      atomicAdd(&seg_sum[(size_t)bb * D + colb + c], local[c]);
    if ((t & 3) == 0) atomicAdd(&seg_cnt[bb], 1.f);
  }
  __syncthreads();
  if (t < D) atomicAdd(&seg_sum[(size_t)b0 * D + t], colsum[t]);
  if (t == 0) atomicAdd(&seg_cnt[b0], scnt);
}

// ---------------- k5: tiny gate-MLP dense layer (B=64, 128x128) ----------------
__global__ void k5_dense(const float* __restrict__ in, const float* __restrict__ W,
                         const float* __restrict__ bias, const float* __restrict__ cnt,
                         float* __restrict__ out, int act) {
  int idx = blockIdx.x * blockDim.x + threadIdx.x;  // over B*D
  int b = idx >> 7;
  int j = idx & 127;
  float scale = cnt ? 1.f / cnt[b] : 1.f;
  float acc = bias[j];
  const float* row = in + (size_t)b * D;
  for (int k = 0; k < D; ++k) acc += row[k] * scale * W[(size_t)k * D + j];
  if (act == 0)
    acc = acc > 0.f ? acc : 0.f;
  else
    acc = 1.f / (1.f + expf(-acc));
  out[idx] = acc;
}

// ---------------- k6: out = (hV + BN2(dh)) * g[batch_id] ----------------
__global__ __launch_bounds__(256) void k6_out(
    const float* __restrict__ hV, const float* __restrict__ dh,
    const int* __restrict__ bid, const float* __restrict__ gbuf,
    const float* __restrict__ bn2_sum, const float* __restrict__ bn2_sq,
    const float* __restrict__ g2, const float* __restrict__ beta2,
    float* __restrict__ out, float invN) {
  __shared__ float cA[D], cB[D];
  int t = threadIdx.x;
  if (t < D) {
    float m = bn2_sum[t] * invN;
    float v = bn2_sq[t] * invN - m * m;
    float A = rsqrtf(v + EPSV) * g2[t];
    cA[t] = A;
    cB[t] = beta2[t] - m * A;
  }
  __syncthreads();
  int base = blockIdx.x * 64;
  int row = base + (t >> 2);
  int colb = (t & 3) * 32;
  int bb = bid[row];
  const float* hv = hV + (size_t)row * D + colb;
  const float* dr = dh + (size_t)row * D + colb;
  const float* gr = gbuf + (size_t)bb * D + colb;
  float* op = out + (size_t)row * D + colb;
#pragma unroll
  for (int c = 0; c < 32; c += 4) {
    float4 a = *(const float4*)(hv + c);
    float4 d = *(const float4*)(dr + c);
    float4 g = *(const float4*)(gr + c);
    float4 o;
    o.x = (a.x + d.x * cA[colb + c + 0] + cB[colb + c + 0]) * g.x;
    o.y = (a.y + d.y * cA[colb + c + 1] + cB[colb + c + 1]) * g.y;
    o.z = (a.z + d.z * cA[colb + c + 2] + cB[colb + c + 2]) * g.z;
    o.w = (a.w + d.w * cA[colb + c + 3] + cB[colb + c + 3]) * g.w;
    *(float4*)(op + c) = o;
  }
}

extern "C" void kernel_launch(void* const* d_in, const int* in_sizes, int n_in,
                              void* d_out, int out_size, void* d_ws, size_t ws_size,
                              hipStream_t stream) {
  const float* hV    = (const float*)d_in[0];
  const int*   bid   = (const int*)d_in[1];
  const float* bn1_g = (const float*)d_in[2];
  const float* bn1_b = (const float*)d_in[3];
  const float* w1    = (const float*)d_in[4];
  const float* b1    = (const float*)d_in[5];
  const float* w2    = (const float*)d_in[6];
  const float* b2    = (const float*)d_in[7];
  const float* bn2_g = (const float*)d_in[8];
  const float* bn2_b = (const float*)d_in[9];
  const float* gw1   = (const float*)d_in[10];
  const float* gb1   = (const float*)d_in[11];
  const float* gw2   = (const float*)d_in[12];
  const float* gb2   = (const float*)d_in[13];
  const float* gw3   = (const float*)d_in[14];
  const float* gb3   = (const float*)d_in[15];
  float* out = (float*)d_out;

  const int N = in_sizes[0] / D;  // 262144
  const float invN = 1.0f / (float)N;

  float* ws = (float*)d_ws;
  float* bn1_sum = ws + 0;
  float* bn1_sq  = ws + 128;
  float* bn2_sum = ws + 256;
  float* bn2_sq  = ws + 384;
  float* bnA     = ws + 512;
  float* bnB     = ws + 640;
  float* segcnt  = ws + 768;
  float* segsum  = ws + 1024;    // 64*128
  float* t1      = ws + 9216;    // 64*128
  float* t2      = ws + 17408;   // 64*128
  float* gbuf    = ws + 25600;   // 64*128
  float* dh_raw  = ws + 34816;   // N*128

  // 1. zero reduction scratch (bn sums, seg_cnt, seg_sum)
  k0_zero<<<36, 256, 0, stream>>>(ws, 9216);
  // 2. BN1 column stats + fold
  k1_bn_stats<<<256, 256, 0, stream>>>(hV, bn1_sum, bn1_sq, N / 256);
  k1b_fold<<<1, 128, 0, stream>>>(bn1_sum, bn1_sq, bn1_g, bn1_b, bnA, bnB, invN);
  // 3. fused dense block (WMMA fp32), BN2 stats accumulated
  k2_fused_mlp<<<N / 128, 256, 0, stream>>>(hV, w1, b1, w2, b2, bnA, bnB,
                                            dh_raw, bn2_sum, bn2_sq);
  // 4. segment sums of h = hV + BN2(dh)
  k4_seg<<<N / 64, 256, 0, stream>>>(hV, dh_raw, bid, bn2_sum, bn2_sq, bn2_g,
                                     bn2_b, segsum, segcnt, invN);
  // 5. gate MLP (c_V = segsum/cnt folded into layer 1)
  k5_dense<<<32, 256, 0, stream>>>(segsum, gw1, gb1, segcnt, t1, 0);
  k5_dense<<<32, 256, 0, stream>>>(t1, gw2, gb2, nullptr, t2, 0);
  k5_dense<<<32, 256, 0, stream>>>(t2, gw3, gb3, nullptr, gbuf, 1);
  // 6. out = h * g[batch_id]
  k6_out<<<N / 64, 256, 0, stream>>>(hV, dh_raw, bid, gbuf, bn2_sum, bn2_sq,
                                     bn2_g, bn2_b, out, invN);
}